// OPTAttention_42314017800961
// MI455X (gfx1250) — compile-verified
//
#include <hip/hip_runtime.h>
#include <hip/hip_bf16.h>

// OPT attention (B=16,T=4,E=2048,H=32,HD=64,P=4096,S=4100), fp32 throughout.
// Memory-bound (~130MB): fp32 WMMA (V_WMMA_F32_16X16X4_F32), split-S flash
// decode for parallelism, GLOBAL_LOAD_ASYNC_TO_LDS_B128 double-buffered K/V.

typedef __attribute__((ext_vector_type(2))) float v2f;
typedef __attribute__((ext_vector_type(8))) float v8f;

#define EDIM 2048
#define HHEADS 32
#define HD 64
#define PP 4096
#define SS 4100
#define TT 4
#define BT 64
#define NSEG 8
#define SEGLEN (PP / NSEG)        // 512 past keys per partial block
#define QSTR 68                   // LDS row stride (floats): 16B-aligned, conflict-free
#define NEG_FLT_MAX (-3.402823466e38f)

#if defined(__HIP_DEVICE_COMPILE__) && __has_builtin(__builtin_amdgcn_global_load_async_to_lds_b128)
#define USE_ASYNC 1
typedef int v4i __attribute__((vector_size(16)));
typedef __attribute__((address_space(1))) v4i gv4i;
typedef __attribute__((address_space(3))) v4i lv4i;
static __device__ __forceinline__ void async_cp16(const float* g, float* l) {
    __builtin_amdgcn_global_load_async_to_lds_b128((gv4i*)g, (lv4i*)l, 0, 0);
}
static __device__ __forceinline__ void wait_async_le4() {
#if __has_builtin(__builtin_amdgcn_s_wait_asynccnt)
    __builtin_amdgcn_s_wait_asynccnt(4);
#else
    asm volatile("s_wait_asynccnt 0x4" ::: "memory");
#endif
}
static __device__ __forceinline__ void wait_async_0() {
#if __has_builtin(__builtin_amdgcn_s_wait_asynccnt)
    __builtin_amdgcn_s_wait_asynccnt(0);
#else
    asm volatile("s_wait_asynccnt 0x0" ::: "memory");
#endif
}
#else
#define USE_ASYNC 0
#endif

static __device__ __forceinline__ v8f wmma4(v2f a, v2f b, v8f c) {
    return __builtin_amdgcn_wmma_f32_16x16x4_f32(false, a, false, b,
                                                 (short)0, c, false, false);
}

// ---------------------------------------------------------------------------
// C[64][2048] = (A[64][2048] @ W[2048][2048]^T + bias) * scale
// One wave per 16x16 tile; grid = (32, 4), block = 128.
// ---------------------------------------------------------------------------
__global__ __launch_bounds__(128) void gemm_xwt(
    const float* __restrict__ A, const float* __restrict__ W,
    const float* __restrict__ bias, float* __restrict__ C, float scale)
{
    const int lane = threadIdx.x & 31;
    const int wave = threadIdx.x >> 5;
    const int half = lane >> 4;
    const int ln   = lane & 15;
    const int row0 = blockIdx.y * 16;
    const int col0 = blockIdx.x * 64 + wave * 16;

    const float* arow = A + (size_t)(row0 + ln) * EDIM;
    const float* wrow = W + (size_t)(col0 + ln) * EDIM;

    v8f c = {};
#pragma unroll 8
    for (int k = 0; k < EDIM; k += 4) {
        v2f a = *(const v2f*)(arow + k + 2 * half);
        v2f b = *(const v2f*)(wrow + k + 2 * half);
        c = wmma4(a, b, c);
    }
    const float bv = bias[col0 + ln];
#pragma unroll
    for (int i = 0; i < 8; ++i) {
        int r = row0 + i + 8 * half;
        C[(size_t)r * EDIM + col0 + ln] = (c[i] + bv) * scale;
    }
}

// ---------------------------------------------------------------------------
// Split-S flash partial: grid = (H=32, NSEG=8), block = 128 (4 waves).
// Each block handles one head x 512 past keys for all 64 query rows and
// writes unnormalized partials (m, l, O) to workspace.
// ---------------------------------------------------------------------------
__global__ __launch_bounds__(128) void attn_partial(
    const float* __restrict__ qws,   // [64][2048], q pre-scaled
    const float* __restrict__ kpast, // [32][4096][64]
    const float* __restrict__ vpast, // [32][4096][64]
    const float* __restrict__ mask,  // [16][4][4100]
    float* __restrict__ pm,          // [32*8*64]
    float* __restrict__ pl,          // [32*8*64]
    float* __restrict__ po)          // [32*8*64][64]
{
    __shared__ float qs[64 * QSTR];
    __shared__ float ks[2][16 * QSTR];
    __shared__ float vs[2][16 * QSTR];
    __shared__ float ps[4][16 * 18];

    const int h    = blockIdx.x;
    const int seg  = blockIdx.y;
    const int tid  = threadIdx.x;
    const int lane = tid & 31;
    const int wave = tid >> 5;
    const int half = lane >> 4;
    const int ln   = lane & 15;
    const int rb   = wave * 16;
    const int sbase = seg * SEGLEN;

    const float* kp = kpast + (size_t)h * PP * HD;
    const float* vp = vpast + (size_t)h * PP * HD;

#if USE_ASYNC
    // Q tile: 64x64 floats = 1024 x b128, 8 per thread.
#pragma unroll
    for (int n = 0; n < 8; ++n) {
        int idx = tid + n * 128;        // [0,1024)
        int r = idx >> 4, dp = idx & 15;
        async_cp16(qws + (size_t)r * EDIM + h * HD + dp * 4, &qs[r * QSTR + dp * 4]);
    }
    // First K/V chunk (4 b128 per thread).
#pragma unroll
    for (int n = 0; n < 2; ++n) {
        int idx = tid + n * 128;        // [0,256)
        int r = idx >> 4, dp = idx & 15;
        async_cp16(kp + (size_t)(sbase + r) * HD + dp * 4, &ks[0][r * QSTR + dp * 4]);
        async_cp16(vp + (size_t)(sbase + r) * HD + dp * 4, &vs[0][r * QSTR + dp * 4]);
    }
#else
    for (int idx = tid; idx < 64 * 64; idx += 128) {
        int r = idx >> 6, d = idx & 63;
        qs[r * QSTR + d] = qws[(size_t)r * EDIM + h * HD + d];
    }
#endif

    float m_i[8], l_i[8];
#pragma unroll
    for (int i = 0; i < 8; ++i) { m_i[i] = NEG_FLT_MAX; l_i[i] = 0.f; }
    v8f o[4] = {};

    const int nchunks = SEGLEN / 16;
    for (int ci = 0; ci < nchunks; ++ci) {
        const int cur = ci & 1;
        const int s0  = sbase + ci * 16;
#if USE_ASYNC
        if (ci + 1 < nchunks) {
            const int s1 = s0 + 16;
#pragma unroll
            for (int n = 0; n < 2; ++n) {
                int idx = tid + n * 128;
                int r = idx >> 4, dp = idx & 15;
                async_cp16(kp + (size_t)(s1 + r) * HD + dp * 4, &ks[cur ^ 1][r * QSTR + dp * 4]);
                async_cp16(vp + (size_t)(s1 + r) * HD + dp * 4, &vs[cur ^ 1][r * QSTR + dp * 4]);
            }
            wait_async_le4();           // only the 4 newest ops still in flight
        } else {
            wait_async_0();
        }
        __syncthreads();
#else
        __syncthreads();
        for (int idx = tid; idx < 16 * 64; idx += 128) {
            int r = idx >> 6, d = idx & 63;
            ks[cur][r * QSTR + d] = kp[(size_t)(s0 + r) * HD + d];
            vs[cur][r * QSTR + d] = vp[(size_t)(s0 + r) * HD + d];
        }
        __syncthreads();
#endif

        // S = Q_tile(16x64) @ K_chunk^T: 16 WMMAs over K=64.
        v8f c = {};
#pragma unroll
        for (int kk = 0; kk < HD; kk += 4) {
            v2f a = *(const v2f*)&qs[(rb + ln) * QSTR + kk + 2 * half];
            v2f b = *(const v2f*)&ks[cur][ln * QSTR + kk + 2 * half];
            c = wmma4(a, b, c);
        }

        // Online softmax update (rows live in 16-lane half-groups, wave32).
        float* psw = ps[wave];
#pragma unroll
        for (int i = 0; i < 8; ++i) {
            const int r  = rb + i + 8 * half;
            const int bb = r >> 2;
            const int t  = r & 3;
            float sv = c[i] + mask[(size_t)(bb * TT + t) * SS + s0 + ln];
            sv = fmaxf(sv, NEG_FLT_MAX);
            float rm = sv;
            rm = fmaxf(rm, __shfl_xor(rm, 1));
            rm = fmaxf(rm, __shfl_xor(rm, 2));
            rm = fmaxf(rm, __shfl_xor(rm, 4));
            rm = fmaxf(rm, __shfl_xor(rm, 8));
            const float mn = fmaxf(m_i[i], rm);
            const float p  = __expf(sv - mn);
            float rs = p;
            rs += __shfl_xor(rs, 1);
            rs += __shfl_xor(rs, 2);
            rs += __shfl_xor(rs, 4);
            rs += __shfl_xor(rs, 8);
            const float corr = __expf(m_i[i] - mn);
            l_i[i] = l_i[i] * corr + rs;
            m_i[i] = mn;
#pragma unroll
            for (int f = 0; f < 4; ++f) o[f][i] *= corr;
            psw[(i + 8 * half) * 18 + ln] = p;
        }

        // O += P(16x16) @ V_chunk(16x64): 16 WMMAs.
#pragma unroll
        for (int kk = 0; kk < 16; kk += 4) {
            v2f a;
            a.x = psw[ln * 18 + kk + 2 * half];
            a.y = psw[ln * 18 + kk + 2 * half + 1];
            const int s1 = kk + 2 * half;
#pragma unroll
            for (int f = 0; f < 4; ++f) {
                v2f b;
                b.x = vs[cur][s1 * QSTR + f * 16 + ln];
                b.y = vs[cur][(s1 + 1) * QSTR + f * 16 + ln];
                o[f] = wmma4(a, b, o[f]);
            }
        }
        __syncthreads();                // chunk fully consumed before overwrite
    }

    // Store unnormalized partials.
#pragma unroll
    for (int i = 0; i < 8; ++i) {
        const int r   = rb + i + 8 * half;
        const int idx = (h * NSEG + seg) * BT + r;
        if (ln == 0) { pm[idx] = m_i[i]; pl[idx] = l_i[i]; }
#pragma unroll
        for (int f = 0; f < 4; ++f)
            po[(size_t)idx * HD + f * 16 + ln] = o[f][i];
    }
}

// ---------------------------------------------------------------------------
// Combine NSEG partials + the 4 batch-dependent new keys.
// grid = (H=32, BT=64), block = 64 (thread = output dim d).
// ---------------------------------------------------------------------------
__global__ __launch_bounds__(64) void attn_reduce(
    const float* __restrict__ qws, const float* __restrict__ kws,
    const float* __restrict__ vws, const float* __restrict__ mask,
    const float* __restrict__ pm, const float* __restrict__ pl,
    const float* __restrict__ po, float* __restrict__ aws)
{
    const int h = blockIdx.x;
    const int r = blockIdx.y;
    const int bb = r >> 2;
    const int t  = r & 3;
    const int d  = threadIdx.x;

    // New-key scores (length-64 dots; redundant across the 64 threads, tiny).
    float sc[4];
#pragma unroll
    for (int j = 0; j < 4; ++j) {
        const float* qrow = qws + (size_t)r * EDIM + h * HD;
        const float* krow = kws + (size_t)(bb * TT + j) * EDIM + h * HD;
        float acc = 0.f;
#pragma unroll 8
        for (int dd = 0; dd < HD; ++dd) acc += qrow[dd] * krow[dd];
        acc += mask[(size_t)(bb * TT + t) * SS + PP + j];
        sc[j] = fmaxf(acc, NEG_FLT_MAX);
    }
    const float mnew = fmaxf(fmaxf(sc[0], sc[1]), fmaxf(sc[2], sc[3]));
    float pnew[4], lnew = 0.f;
#pragma unroll
    for (int j = 0; j < 4; ++j) { pnew[j] = __expf(sc[j] - mnew); lnew += pnew[j]; }
    float onew = 0.f;
#pragma unroll
    for (int j = 0; j < 4; ++j)
        onew += pnew[j] * vws[(size_t)(bb * TT + j) * EDIM + h * HD + d];

    // Global max across segments.
    float M = mnew;
#pragma unroll
    for (int s = 0; s < NSEG; ++s)
        M = fmaxf(M, pm[(h * NSEG + s) * BT + r]);

    float wnew = __expf(mnew - M);
    float L = lnew * wnew;
    float O = onew * wnew;
#pragma unroll
    for (int s = 0; s < NSEG; ++s) {
        const int idx = (h * NSEG + s) * BT + r;
        const float w = __expf(pm[idx] - M);
        L += pl[idx] * w;
        O += po[(size_t)idx * HD + d] * w;
    }
    aws[(size_t)r * EDIM + h * HD + d] = O / L;
}

// ---------------------------------------------------------------------------
extern "C" void kernel_launch(void* const* d_in, const int* in_sizes, int n_in,
                              void* d_out, int out_size, void* d_ws, size_t ws_size,
                              hipStream_t stream) {
    const float* x     = (const float*)d_in[0];
    const float* kpast = (const float*)d_in[1];
    const float* vpast = (const float*)d_in[2];
    const float* mask  = (const float*)d_in[3];
    const float* Wq    = (const float*)d_in[4];
    const float* bq    = (const float*)d_in[5];
    const float* Wk    = (const float*)d_in[6];
    const float* bk    = (const float*)d_in[7];
    const float* Wv    = (const float*)d_in[8];
    const float* bv    = (const float*)d_in[9];
    const float* Wo    = (const float*)d_in[10];
    const float* bo    = (const float*)d_in[11];
    float* out = (float*)d_out;

    // Workspace layout (floats): qws,kws,vws,aws [64][2048] each,
    // pm/pl [32*8*64], po [32*8*64][64]  -> ~6.2 MB total.
    float* ws  = (float*)d_ws;
    float* qws = ws;
    float* kws = qws + (size_t)BT * EDIM;
    float* vws = kws + (size_t)BT * EDIM;
    float* aws = vws + (size_t)BT * EDIM;
    float* pm  = aws + (size_t)BT * EDIM;
    float* pl  = pm + (size_t)HHEADS * NSEG * BT;
    float* po  = pl + (size_t)HHEADS * NSEG * BT;

    const float SCALE = 0.125f;  // 64^-0.5

    dim3 gg(EDIM / 64, BT / 16);  // (32, 4)
    gemm_xwt<<<gg, 128, 0, stream>>>(x, Wq, bq, qws, SCALE);
    gemm_xwt<<<gg, 128, 0, stream>>>(x, Wk, bk, kws, 1.0f);
    gemm_xwt<<<gg, 128, 0, stream>>>(x, Wv, bv, vws, 1.0f);

    attn_partial<<<dim3(HHEADS, NSEG), 128, 0, stream>>>(
        qws, kpast, vpast, mask, pm, pl, po);
    attn_reduce<<<dim3(HHEADS, BT), 64, 0, stream>>>(
        qws, kws, vws, mask, pm, pl, po, aws);

    gemm_xwt<<<gg, 128, 0, stream>>>(aws, Wo, bo, out, 1.0f);
}